// dynamic_resnetXt_covariance_23596550324487
// MI455X (gfx1250) — compile-verified
//
#include <hip/hip_runtime.h>
#include <hip/hip_bf16.h>

typedef __attribute__((ext_vector_type(2))) float v2f;
typedef __attribute__((ext_vector_type(8))) float v8f;

#define NPTS 2048
#define NB   8
#define NK   40

// ---------------------------------------------------------------------------
// Squared norms xx[b*N+n] = sum_c x[b,c,n]^2
// ---------------------------------------------------------------------------
__global__ void k_xx(const float* __restrict__ x, float* __restrict__ xx, int N) {
  int i = blockIdx.x * blockDim.x + threadIdx.x;   // i = b*N + n
  int b = i / N, n = i - b * N;
  const float* xb = x + (size_t)b * 3 * N;
  float v0 = xb[n], v1 = xb[N + n], v2 = xb[2 * N + n];
  xx[i] = v0 * v0 + v1 * v1 + v2 * v2;
}

// ---------------------------------------------------------------------------
// kNN: top-40 of pd[m] = 2*<x_n,x_m> - xx_n - xx_m, per (b,n). One block per row.
// ---------------------------------------------------------------------------
__global__ __launch_bounds__(128) void k_knn(const float* __restrict__ x,
                                             const float* __restrict__ xx,
                                             int* __restrict__ idxb, int N) {
  __shared__ float d[NPTS];
  __shared__ float rv[128];
  __shared__ int   ri[128];
  int tid = threadIdx.x;
  int bn = blockIdx.x;
  int b = bn / N, n = bn - b * N;
  const float* xb = x + (size_t)b * 3 * N;
  float x0 = xb[n], x1 = xb[N + n], x2 = xb[2 * N + n];
  float xxn = xx[bn];
  for (int m = tid; m < N; m += 128) {
    float dot = x0 * xb[m] + x1 * xb[N + m] + x2 * xb[2 * N + m];
    d[m] = 2.0f * dot - xxn - xx[b * N + m];
  }
  __syncthreads();
  for (int r = 0; r < NK; ++r) {
    float bv = -3.4e38f; int bi = N - 1;
    for (int m = tid; m < N; m += 128) {
      float v = d[m];
      if (v > bv) { bv = v; bi = m; }   // ascending m per thread -> lowest idx kept
    }
    rv[tid] = bv; ri[tid] = bi;
    __syncthreads();
    for (int s = 64; s > 0; s >>= 1) {
      if (tid < s) {
        float ov = rv[tid + s]; int oi = ri[tid + s];
        if (ov > rv[tid] || (ov == rv[tid] && oi < ri[tid])) { rv[tid] = ov; ri[tid] = oi; }
      }
      __syncthreads();
    }
    if (tid == 0) { idxb[bn * NK + r] = ri[0]; d[ri[0]] = -3.4e38f; }
    __syncthreads();
  }
}

// ---------------------------------------------------------------------------
// Batch-mean of rel: mean[n,k,c] = (1/B) sum_b ( x[b,c,idx[b,n,k]] - x[b,c,n] )
// ---------------------------------------------------------------------------
__global__ void k_mean(const float* __restrict__ x, const int* __restrict__ idxb,
                       float* __restrict__ meanr, int N) {
  int i = blockIdx.x * blockDim.x + threadIdx.x;   // i = (n*K + k)*3 + c
  if (i >= N * NK * 3) return;
  int c = i % 3;
  int k = (i / 3) % NK;
  int n = i / (3 * NK);
  float s = 0.f;
  for (int b = 0; b < NB; ++b) {
    int j = idxb[((size_t)b * N + n) * NK + k];
    const float* xc = x + ((size_t)b * 3 + c) * N;
    s += xc[j] - xc[n];
  }
  meanr[i] = s * (1.0f / NB);
}

// ---------------------------------------------------------------------------
// Fused edge stage per (b,n): rel, var(ddof=1), relf, norm, cov(40x40),
// conv W3->W4 (rel branch) and W5->W6 (cov branch) with lrelu, max over K.
// Writes channels [0,128) of the a1 concat buffer (B,256,N).
// ---------------------------------------------------------------------------
__global__ __launch_bounds__(64) void k_edge(
    const float* __restrict__ x, const int* __restrict__ idxb,
    const float* __restrict__ meanr,
    const float* __restrict__ W3, const float* __restrict__ s3, const float* __restrict__ b3,
    const float* __restrict__ W4, const float* __restrict__ s4, const float* __restrict__ b4,
    const float* __restrict__ W5, const float* __restrict__ s5, const float* __restrict__ b5,
    const float* __restrict__ W6, const float* __restrict__ s6, const float* __restrict__ b6,
    float* __restrict__ outA1, int N) {
  __shared__ float rel[NK * 3], nrm[NK * 3], relf[NK * 3];
  __shared__ float xp[3], vmul[3];
  __shared__ float cov[NK * NK];
  __shared__ float hA[NK * 64], hB[NK * 64];
  int tid = threadIdx.x;
  int bn = blockIdx.x;
  int b = bn / N, n = bn - b * N;
  const float* xb = x + (size_t)b * 3 * N;
  if (tid < 3) xp[tid] = xb[tid * N + n];
  __syncthreads();
  if (tid < NK) {
    int j = idxb[bn * NK + tid];
    for (int c = 0; c < 3; ++c) rel[tid * 3 + c] = xb[c * N + j] - xp[c];
  }
  __syncthreads();
  if (tid < 3) {
    float mu = 0.f;
    for (int k = 0; k < NK; ++k) mu += rel[k * 3 + tid];
    mu *= (1.f / NK);
    float v = 0.f;
    for (int k = 0; k < NK; ++k) { float dd = rel[k * 3 + tid] - mu; v += dd * dd; }
    vmul[tid] = 1.f - v * (1.f / (NK - 1));
  }
  __syncthreads();
  for (int t = tid; t < NK * 3; t += 64) {
    int c = t % 3; int k = t / 3;
    relf[t] = rel[t] * vmul[c];
    nrm[t]  = rel[t] - meanr[(n * NK + k) * 3 + c];
  }
  __syncthreads();
  for (int t = tid; t < NK * NK; t += 64) {
    int k = t / NK, j = t - k * NK;
    cov[t] = rel[k * 3] * nrm[j * 3] + rel[k * 3 + 1] * nrm[j * 3 + 1] + rel[k * 3 + 2] * nrm[j * 3 + 2];
  }
  __syncthreads();
  // ---- rel branch: H3 = lrelu(s3*(F_r W3^T)+b3) ----
  for (int t = tid; t < NK * 64; t += 64) {
    int k = t >> 6, o = t & 63;
    const float* w = W3 + o * 6;
    float a = xp[0] * w[0] + xp[1] * w[1] + xp[2] * w[2]
            + relf[k * 3] * w[3] + relf[k * 3 + 1] * w[4] + relf[k * 3 + 2] * w[5];
    a = a * s3[o] + b3[o];
    hA[t] = a > 0.f ? a : 0.2f * a;
  }
  __syncthreads();
  for (int t = tid; t < NK * 64; t += 64) {
    int k = t >> 6, o = t & 63;
    const float* w = W4 + o * 64;
    const float* h = hA + k * 64;
    float a = 0.f;
    for (int c = 0; c < 64; ++c) a += h[c] * w[c];
    a = a * s4[o] + b4[o];
    hB[t] = a > 0.f ? a : 0.2f * a;
  }
  __syncthreads();
  {
    float m = -3.4e38f;
    for (int k = 0; k < NK; ++k) m = fmaxf(m, hB[k * 64 + tid]);
    outA1[(size_t)b * 256 * N + (size_t)tid * N + n] = m;
  }
  __syncthreads();
  // ---- cov branch: H5 = lrelu(s5*(F_c W5^T)+b5) ----
  for (int t = tid; t < NK * 64; t += 64) {
    int k = t >> 6, o = t & 63;
    const float* w = W5 + o * 43;
    float a = xp[0] * w[0] + xp[1] * w[1] + xp[2] * w[2];
    const float* ck = cov + k * NK;
    for (int j = 0; j < NK; ++j) a += ck[j] * w[3 + j];
    a = a * s5[o] + b5[o];
    hA[t] = a > 0.f ? a : 0.2f * a;
  }
  __syncthreads();
  for (int t = tid; t < NK * 64; t += 64) {
    int k = t >> 6, o = t & 63;
    const float* w = W6 + o * 64;
    const float* h = hA + k * 64;
    float a = 0.f;
    for (int c = 0; c < 64; ++c) a += h[c] * w[c];
    a = a * s6[o] + b6[o];
    hB[t] = a > 0.f ? a : 0.2f * a;
  }
  __syncthreads();
  {
    float m = -3.4e38f;
    for (int k = 0; k < NK; ++k) m = fmaxf(m, hB[k * 64 + tid]);
    outA1[(size_t)b * 256 * N + (size_t)(64 + tid) * N + n] = m;
  }
}

// ---------------------------------------------------------------------------
// Generic per-point GEMM via V_WMMA_F32_16X16X4_F32.
// X: (B, Cstride, N) channel-major, channels [0,Cin), Cin % 4 == 0.
// W: (O, Wld) row-major.  Y[b,o,n] = epi( sum_c X[b,c,n]*W[o,c] ),
// epi = lrelu( scale*( . + bias2[b,o] ) + bias ).
// Each wave: 4 M-tiles (64 points) x 16 outputs -> 4 WMMAs per k-step sharing
// one 64-bit W load. Block = 4 waves = 64 points x 64 outputs.
// MaxOut != null: atomic max-reduce over n into monotone-uint keys (B,O).
// Out-of-range output columns are handled by clamping the W row pointer:
// garbage feeds only the discarded D columns (WMMA column n is fed solely by
// lane n's B data), writes are guarded.
// ---------------------------------------------------------------------------
__device__ __forceinline__ unsigned f32key(float f) {
  unsigned u = __float_as_uint(f);
  return (u & 0x80000000u) ? ~u : (u | 0x80000000u);
}

__global__ __launch_bounds__(128) void k_gemm(
    const float* __restrict__ X, const float* __restrict__ W,
    const float* __restrict__ S, const float* __restrict__ Bv,
    const float* __restrict__ Bias2, float* __restrict__ Y,
    unsigned int* __restrict__ MaxOut,
    int Cin, int Cstride, int Wld, int O, int Ostride, int N, int act) {
  int lane = threadIdx.x & 31;
  int wave = threadIdx.x >> 5;
  int row0 = blockIdx.x << 6;                 // 64 points per block
  int col0 = blockIdx.y * 64 + wave * 16;     // 16 outputs per wave
  int b = row0 / N;
  int n0 = row0 - b * N;
  int m  = lane & 15;
  int hi = lane >> 4;
  const float* Xb = X + (size_t)b * Cstride * N + n0 + m;
  int o = col0 + m;
  const float* Wo = W + (size_t)(o < O ? o : O - 1) * Wld;
  v8f acc[4];
  acc[0] = v8f{0.f,0.f,0.f,0.f,0.f,0.f,0.f,0.f};
  acc[1] = acc[0]; acc[2] = acc[0]; acc[3] = acc[0];
#pragma unroll 2
  for (int k = 0; k < Cin; k += 4) {
    int ka = k + hi * 2;                       // f32 16x4 A / 4x16 B lane split
    const float* xk = Xb + (size_t)ka * N;     // K = ka row
    const float* xk1 = xk + N;                 // K = ka+1 row
    v2f bf  = { Wo[ka], Wo[ka + 1] };          // contiguous -> b64 load
    v2f af0 = { xk[0],  xk1[0]  };
    v2f af1 = { xk[16], xk1[16] };
    v2f af2 = { xk[32], xk1[32] };
    v2f af3 = { xk[48], xk1[48] };
    acc[0] = __builtin_amdgcn_wmma_f32_16x16x4_f32(false, af0, false, bf, (short)0, acc[0], false, false);
    acc[1] = __builtin_amdgcn_wmma_f32_16x16x4_f32(false, af1, false, bf, (short)0, acc[1], false, false);
    acc[2] = __builtin_amdgcn_wmma_f32_16x16x4_f32(false, af2, false, bf, (short)0, acc[2], false, false);
    acc[3] = __builtin_amdgcn_wmma_f32_16x16x4_f32(false, af3, false, bf, (short)0, acc[3], false, false);
  }
  if (o >= O) return;
  float s  = S     ? S[o]             : 1.f;
  float bb = Bv    ? Bv[o]            : 0.f;
  float g2 = Bias2 ? Bias2[b * O + o] : 0.f;
  if (MaxOut) {
    float mx = -3.4e38f;
#pragma unroll
    for (int t = 0; t < 4; ++t)
#pragma unroll
      for (int i = 0; i < 8; ++i) {
        float v = (acc[t][i] + g2) * s + bb;
        if (act) v = v > 0.f ? v : 0.2f * v;
        mx = fmaxf(mx, v);
      }
    atomicMax(&MaxOut[b * O + o], f32key(mx));
  } else {
    float* Yb = Y + (size_t)b * Ostride * N + (size_t)o * N + n0 + hi * 8;
#pragma unroll
    for (int t = 0; t < 4; ++t) {
      float* Yp = Yb + t * 16;   // D: VGPR i holds row M = i + 8*hi of tile t
#pragma unroll
      for (int i = 0; i < 8; ++i) {
        float v = (acc[t][i] + g2) * s + bb;
        if (act) v = v > 0.f ? v : 0.2f * v;
        Yp[i] = v;
      }
    }
  }
}

// ---------------------------------------------------------------------------
// Small helpers
// ---------------------------------------------------------------------------
__global__ void k_zero_u32(unsigned* __restrict__ p, int n) {
  int i = blockIdx.x * blockDim.x + threadIdx.x;
  if (i < n) p[i] = 0u;
}

__global__ void k_copy256(const float* __restrict__ a1, float* __restrict__ a2, int N) {
  int i = blockIdx.x * blockDim.x + threadIdx.x;
  int per = 256 * N;
  if (i >= NB * per) return;
  int b = i / per, r = i - b * per;
  a2[(size_t)b * 512 * N + r] = a1[i];
}

// g vector (B,1088): decode max keys (1024) + lf = lrelu(s8*(W8 l)+b8) (64)
__global__ void k_gvec(const unsigned* __restrict__ gkeys, const float* __restrict__ l,
                       const float* __restrict__ W8, const float* __restrict__ s8,
                       const float* __restrict__ b8, float* __restrict__ gvec) {
  int i = blockIdx.x * blockDim.x + threadIdx.x;
  if (i >= NB * 1088) return;
  int b = i / 1088, c = i - b * 1088;
  float g;
  if (c < 1024) {
    unsigned key = gkeys[b * 1024 + c];
    unsigned u = (key & 0x80000000u) ? (key & 0x7FFFFFFFu) : ~key;
    g = __uint_as_float(u);
  } else {
    int o = c - 1024;
    const float* w = W8 + o * 16;
    float a = 0.f;
    for (int j = 0; j < 16; ++j) a += w[j] * l[b * 16 + j];
    a = a * s8[o] + b8[o];
    g = a > 0.f ? a : 0.2f * a;
  }
  gvec[i] = g;
}

// g9[b,o] = sum_{c<1088} W9[o,c] * g[b,c]  (folds the broadcast-g half of W9)
__global__ void k_g9(const float* __restrict__ W9, const float* __restrict__ gvec,
                     float* __restrict__ g9) {
  int i = blockIdx.x * blockDim.x + threadIdx.x;
  if (i >= NB * 512) return;
  int b = i / 512, o = i - b * 512;
  const float* w = W9 + (size_t)o * 1600;
  const float* g = gvec + b * 1088;
  float a = 0.f;
  for (int c = 0; c < 1088; ++c) a += w[c] * g[c];
  g9[i] = a;
}

// ---------------------------------------------------------------------------
extern "C" void kernel_launch(void* const* d_in, const int* in_sizes, int n_in,
                              void* d_out, int out_size, void* d_ws, size_t ws_size,
                              hipStream_t stream) {
  (void)in_sizes; (void)n_in; (void)out_size; (void)ws_size;
  const int N = NPTS, BN = NB * NPTS;

  const float* x = (const float*)d_in[0];
  const float* l = (const float*)d_in[1];
  const float *W3w = (const float*)d_in[2],  *s3v = (const float*)d_in[3],  *b3v = (const float*)d_in[4];
  const float *W4w = (const float*)d_in[5],  *s4v = (const float*)d_in[6],  *b4v = (const float*)d_in[7];
  const float *W5w = (const float*)d_in[8],  *s5v = (const float*)d_in[9],  *b5v = (const float*)d_in[10];
  const float *W6w = (const float*)d_in[11], *s6v = (const float*)d_in[12], *b6v = (const float*)d_in[13];
  const float *We1a = (const float*)d_in[14], *se1a = (const float*)d_in[15], *be1a = (const float*)d_in[16];
  const float *We1b = (const float*)d_in[17], *se1b = (const float*)d_in[18], *be1b = (const float*)d_in[19];
  const float *We2a = (const float*)d_in[20], *se2a = (const float*)d_in[21], *be2a = (const float*)d_in[22];
  const float *We2b = (const float*)d_in[23], *se2b = (const float*)d_in[24], *be2b = (const float*)d_in[25];
  const float *We3  = (const float*)d_in[26], *se3  = (const float*)d_in[27], *be3  = (const float*)d_in[28];
  const float *W7w  = (const float*)d_in[29], *s7v  = (const float*)d_in[30], *b7v  = (const float*)d_in[31];
  const float *W8w  = (const float*)d_in[32], *s8v  = (const float*)d_in[33], *b8v  = (const float*)d_in[34];
  const float *W9w  = (const float*)d_in[35], *s9v  = (const float*)d_in[36], *b9v  = (const float*)d_in[37];
  const float *W10w = (const float*)d_in[38], *s10v = (const float*)d_in[39], *b10v = (const float*)d_in[40];
  const float *W11w = (const float*)d_in[41], *s11v = (const float*)d_in[42], *b11v = (const float*)d_in[43];
  const float *W12w = (const float*)d_in[44];

  // ---- workspace layout (256B aligned) ----
  char* ws = (char*)d_ws;
  size_t off = 0;
  auto alloc = [&](size_t bytes) -> void* {
    off = (off + 255) & ~(size_t)255;
    void* p = ws + off;
    off += bytes;
    return p;
  };
  float*    xxb    = (float*)alloc((size_t)BN * 4);                 // (B,N)
  int*      idxb   = (int*)alloc((size_t)BN * NK * 4);              // (B,N,K)
  float*    meanr  = (float*)alloc((size_t)N * NK * 3 * 4);         // (N,K,3)
  float*    a1     = (float*)alloc((size_t)NB * 256 * N * 4);       // (B,256,N)
  float*    tmp128 = (float*)alloc((size_t)NB * 128 * N * 4);       // (B,128,N)
  float*    a2     = (float*)alloc((size_t)NB * 512 * N * 4);       // (B,512,N)
  float*    tmp256 = (float*)alloc((size_t)NB * 256 * N * 4);       // (B,256,N)
  float*    enc    = (float*)alloc((size_t)NB * 512 * N * 4);       // (B,512,N)
  unsigned* gkeys  = (unsigned*)alloc((size_t)NB * 1024 * 4);       // (B,1024)
  float*    gvec   = (float*)alloc((size_t)NB * 1088 * 4);          // (B,1088)
  float*    g9     = (float*)alloc((size_t)NB * 512 * 4);           // (B,512)
  float*    h9  = a2;      // reuse: a2 dead after e3
  float*    h10 = tmp256;  // reuse
  float*    h11 = tmp128;  // reuse

  // 1) squared norms
  k_xx<<<BN / 256, 256, 0, stream>>>(x, xxb, N);
  // 2) kNN top-40
  k_knn<<<BN, 128, 0, stream>>>(x, xxb, idxb, N);
  // 3) batch mean of rel
  k_mean<<<(N * NK * 3 + 255) / 256, 256, 0, stream>>>(x, idxb, meanr, N);
  // 4) fused edge stage -> a1 channels [0,128)
  k_edge<<<BN, 64, 0, stream>>>(x, idxb, meanr,
                                W3w, s3v, b3v, W4w, s4v, b4v,
                                W5w, s5v, b5v, W6w, s6v, b6v, a1, N);
  dim3 blk(128);
  const int GX = BN / 64;   // 256 blocks of 64 points
  // 5) e1a: a1[0:128] -> tmp128
  k_gemm<<<dim3(GX, 2), blk, 0, stream>>>(a1, We1a, se1a, be1a, nullptr,
      tmp128, nullptr, 128, 256, 128, 128, 128, N, 1);
  // 6) e1b: tmp128 -> a1 channels [128,256)
  k_gemm<<<dim3(GX, 2), blk, 0, stream>>>(tmp128, We1b, se1b, be1b, nullptr,
      a1 + (size_t)128 * N, nullptr, 128, 128, 128, 128, 256, N, 1);
  // 7) a2[0:256] = a1
  k_copy256<<<(NB * 256 * N + 255) / 256, 256, 0, stream>>>(a1, a2, N);
  // 8) e2a: a2[0:256] -> tmp256
  k_gemm<<<dim3(GX, 4), blk, 0, stream>>>(a2, We2a, se2a, be2a, nullptr,
      tmp256, nullptr, 256, 512, 256, 256, 256, N, 1);
  // 9) e2b: tmp256 -> a2 channels [256,512)
  k_gemm<<<dim3(GX, 4), blk, 0, stream>>>(tmp256, We2b, se2b, be2b, nullptr,
      a2 + (size_t)256 * N, nullptr, 256, 256, 256, 256, 512, N, 1);
  // 10) e3: a2 -> enc
  k_gemm<<<dim3(GX, 8), blk, 0, stream>>>(a2, We3, se3, be3, nullptr,
      enc, nullptr, 512, 512, 512, 512, 512, N, 1);
  // 11) init global-max keys
  k_zero_u32<<<(NB * 1024 + 255) / 256, 256, 0, stream>>>(gkeys, NB * 1024);
  // 12) W7 + max over n -> gkeys
  k_gemm<<<dim3(GX, 16), blk, 0, stream>>>(enc, W7w, s7v, b7v, nullptr,
      nullptr, gkeys, 512, 512, 512, 1024, 0, N, 1);
  // 13) g vector (decode max, append lf)
  k_gvec<<<(NB * 1088 + 127) / 128, 128, 0, stream>>>(gkeys, l, W8w, s8v, b8v, gvec);
  // 14) fold g half of W9
  k_g9<<<(NB * 512 + 127) / 128, 128, 0, stream>>>(W9w, gvec, g9);
  // 15) W9: enc half (W9 cols [1088,1600)) + g9 pre-bias -> h9
  k_gemm<<<dim3(GX, 8), blk, 0, stream>>>(enc, W9w + 1088, s9v, b9v, g9,
      h9, nullptr, 512, 512, 1600, 512, 512, N, 1);
  // 16) W10: h9 -> h10
  k_gemm<<<dim3(GX, 4), blk, 0, stream>>>(h9, W10w, s10v, b10v, nullptr,
      h10, nullptr, 512, 512, 512, 256, 256, N, 1);
  // 17) W11: h10 -> h11
  k_gemm<<<dim3(GX, 2), blk, 0, stream>>>(h10, W11w, s11v, b11v, nullptr,
      h11, nullptr, 256, 256, 256, 128, 128, N, 1);
  // 18) W12: h11 -> out (B,50,N), no scale/bias/act
  k_gemm<<<dim3(GX, 1), blk, 0, stream>>>(h11, W12w, nullptr, nullptr, nullptr,
      (float*)d_out, nullptr, 128, 128, 128, 50, 50, N, 0);
}